// ConvexOptimExtractor_25417616458292
// MI455X (gfx1250) — compile-verified
//
#include <hip/hip_runtime.h>
#include <hip/hip_bf16.h>
#include <math.h>

typedef float v2f __attribute__((ext_vector_type(2)));
typedef float v4f __attribute__((ext_vector_type(4)));
typedef float v8f __attribute__((ext_vector_type(8)));
typedef int v4i __attribute__((vector_size(4 * sizeof(int))));

#define D_DIM 768
#define M_DIM 3072
#define B_DIM 128
#define REG_L 0.1f
#define NUM_ITER 500
#define HIST 300.0f
#define LOWEST 0.001f
#define STEP 0.005f
#define POWER_ITERS 50

// ---- CDNA5 async global->LDS copy (ASYNCcnt-tracked), with sync fallback -----
#if defined(__has_builtin)
#if __has_builtin(__builtin_amdgcn_global_load_async_to_lds_b128)
#define ASYNC_LDS 1
#endif
#endif
#ifndef ASYNC_LDS
#define ASYNC_LDS 0
#endif

#if ASYNC_LDS
typedef __attribute__((address_space(1))) v4i* gv4i_p;  // global int4*
typedef __attribute__((address_space(3))) v4i* lv4i_p;  // LDS int4*
#endif

static __device__ __forceinline__ void cp16(float* lds, const float* g) {
#if ASYNC_LDS
  __builtin_amdgcn_global_load_async_to_lds_b128((gv4i_p)g, (lv4i_p)lds, 0, 0);
#else
  v4f v = *(const v4f*)g;
  lds[0] = v.x; lds[1] = v.y; lds[2] = v.z; lds[3] = v.w;
#endif
}

static __device__ __forceinline__ void async_fence() {
#if ASYNC_LDS
  asm volatile("s_wait_asynccnt 0x0" ::: "memory");
#endif
}

static __device__ __forceinline__ v8f wmma4(v2f a, v2f b, v8f c) {
  // D = A(16x4 f32) * B(4x16 f32) + C(16x16 f32), wave32
  return __builtin_amdgcn_wmma_f32_16x16x4_f32(false, a, false, b, (short)0, c,
                                               false, false);
}

// ---------------------------------------------------------------- init / coeffs
__global__ void k_init(float* __restrict__ Yv, float* __restrict__ ahat,
                       float* __restrict__ vvec, float* __restrict__ scal) {
  int idx = blockIdx.x * blockDim.x + threadIdx.x;
  if (idx < M_DIM * B_DIM) { Yv[idx] = 0.f; ahat[idx] = 0.f; }
  if (idx < D_DIM) vvec[idx] = 1.0f;
  if (idx == 0) {
    scal[4] = 0.f;                         // signal energy acc
    scal[5] = 0.f;                         // noise energy acc
    ((unsigned*)scal)[6] = 0u;             // max bits (nonneg floats)
    ((unsigned*)scal)[7] = 0x7f800000u;    // min bits = +inf
  }
}

__global__ void k_coeffs(float* __restrict__ c) {
  if (threadIdx.x == 0 && blockIdx.x == 0) {
    float tk = 1.f;
    for (int i = 0; i < NUM_ITER; ++i) {
      float tkn = (1.f + sqrtf(1.f + 4.f * tk * tk)) * 0.5f;
      c[i] = (tk - 1.f) / tkn;
      tk = tkn;
    }
  }
}

// ------------------------------------------------- power iteration for L = eigmax
__global__ void k_pk_a(const float* __restrict__ PHI, const float* __restrict__ vvec,
                       float* __restrict__ tvec, float* __restrict__ scal) {
  int m = blockIdx.x * blockDim.x + threadIdx.x;
  if (m == 0) scal[0] = 0.f;  // zero norm accumulator for following pk_b
  if (m < M_DIM) {
    float s = 0.f;
    for (int d = 0; d < D_DIM; ++d) s = fmaf(PHI[d * M_DIM + m], vvec[d], s);
    tvec[m] = s;
  }
}

__global__ void k_pk_b(const float* __restrict__ PHI, float* __restrict__ vvec,
                       const float* __restrict__ tvec, float* __restrict__ scal) {
  __shared__ float red[256];
  int d = blockIdx.x * blockDim.x + threadIdx.x;
  float s = 0.f;
  if (d < D_DIM) {
    for (int m = 0; m < M_DIM; ++m) s = fmaf(PHI[d * M_DIM + m], tvec[m], s);
    vvec[d] = s;
  }
  red[threadIdx.x] = s * s;
  __syncthreads();
  for (int st = 128; st > 0; st >>= 1) {
    if (threadIdx.x < st) red[threadIdx.x] += red[threadIdx.x + st];
    __syncthreads();
  }
  if (threadIdx.x == 0) atomicAdd(&scal[0], red[0]);
}

__global__ void k_pk_fin(float* __restrict__ vvec, float* __restrict__ scal) {
  float L = sqrtf(scal[0]);
  float inv = (L > 0.f) ? 1.f / L : 0.f;
  int d = threadIdx.x;
  if (d < D_DIM) vvec[d] *= inv;
  if (d == 0) {
    scal[1] = L;            // lambda_max estimate
    scal[2] = inv;          // eta = 1/L
    scal[3] = REG_L * inv;  // eta * lambda (threshold)
  }
}

// ----------------------------------------------- GEMM: Res = I - PHI * Y  (K = M)
// block = 128 threads (4 waves); tile = 64 d-rows x 16 b-cols, one 16x16 per wave
// double-buffered LDS panels, async global->LDS staging overlapped with WMMA
__global__ __launch_bounds__(128) void k_gemm_res(
    const float* __restrict__ PHI, const float* __restrict__ Y,
    const float* __restrict__ batch, float* __restrict__ Res,
    float* __restrict__ recons) {
  __shared__ __align__(16) float ldsA[2][64 * 68];  // PHI panel [64 d][64 k]
  __shared__ __align__(16) float ldsY[2][64 * 20];  // Y panel   [64 k][16 b]
  const int tid = threadIdx.x;
  const int lane = tid & 31, wave = tid >> 5;
  const int half = lane >> 4, l16 = lane & 15;
  const int dblk = blockIdx.x * 64;
  const int b0 = blockIdx.y * 16;
  v8f acc = {};

  // prologue: stage chunk 0 into buffer 0
  for (int i = tid; i < 64 * 16; i += 128) {
    int r = i >> 4, c4 = (i & 15) << 2;
    cp16(&ldsA[0][r * 68 + c4], &PHI[(dblk + r) * M_DIM + c4]);
  }
  for (int i = tid; i < 64 * 4; i += 128) {
    int r = i >> 2, c4 = (i & 3) << 2;
    cp16(&ldsY[0][r * 20 + c4], &Y[r * B_DIM + b0 + c4]);
  }
  async_fence();
  __syncthreads();

  int cur = 0;
  for (int kk = 0; kk < M_DIM; kk += 64) {
    const int nxt = cur ^ 1;
    if (kk + 64 < M_DIM) {  // issue async stage of next chunk before compute
      for (int i = tid; i < 64 * 16; i += 128) {
        int r = i >> 4, c4 = (i & 15) << 2;
        cp16(&ldsA[nxt][r * 68 + c4], &PHI[(dblk + r) * M_DIM + kk + 64 + c4]);
      }
      for (int i = tid; i < 64 * 4; i += 128) {
        int r = i >> 2, c4 = (i & 3) << 2;
        cp16(&ldsY[nxt][r * 20 + c4], &Y[(kk + 64 + r) * B_DIM + b0 + c4]);
      }
    }
    const float* Ap = ldsA[cur];
    const float* Yp = ldsY[cur];
#pragma unroll
    for (int k4 = 0; k4 < 64; k4 += 4) {
      int kb = k4 + half * 2;
      v2f a = *(const v2f*)&Ap[(wave * 16 + l16) * 68 + kb];
      v2f b;
      b.x = Yp[kb * 20 + l16];
      b.y = Yp[(kb + 1) * 20 + l16];
      acc = wmma4(a, b, acc);
    }
    async_fence();
    __syncthreads();
    cur = nxt;
  }

  const int b = b0 + l16;
#pragma unroll
  for (int i = 0; i < 8; ++i) {
    int d = dblk + wave * 16 + half * 8 + i;
    float r = batch[b * D_DIM + d] - acc[i];  // I[d,b] = batch[b,d]
    Res[d * B_DIM + b] = r;
    if (recons) recons[d * B_DIM + b] = acc[i];
  }
}

// --------------------------- GEMM: g = PHI^T * Res (K = D) + FISTA state update
__global__ __launch_bounds__(128) void k_gemm_grad(
    const float* __restrict__ PHI, const float* __restrict__ Res,
    float* __restrict__ ahat, float* __restrict__ Yv,
    const float* __restrict__ coeffs, const float* __restrict__ scal, int it) {
  __shared__ __align__(16) float ldsAT[2][64 * 68];  // PHI panel [64 d][64 m]
  __shared__ __align__(16) float ldsB[2][64 * 20];   // Res panel [64 d][16 b]
  const int tid = threadIdx.x;
  const int lane = tid & 31, wave = tid >> 5;
  const int half = lane >> 4, l16 = lane & 15;
  const int mblk = blockIdx.x * 64;
  const int b0 = blockIdx.y * 16;
  v8f acc = {};

  for (int i = tid; i < 64 * 16; i += 128) {
    int r = i >> 4, c4 = (i & 15) << 2;
    cp16(&ldsAT[0][r * 68 + c4], &PHI[r * M_DIM + mblk + c4]);
  }
  for (int i = tid; i < 64 * 4; i += 128) {
    int r = i >> 2, c4 = (i & 3) << 2;
    cp16(&ldsB[0][r * 20 + c4], &Res[r * B_DIM + b0 + c4]);
  }
  async_fence();
  __syncthreads();

  int cur = 0;
  for (int kk = 0; kk < D_DIM; kk += 64) {
    const int nxt = cur ^ 1;
    if (kk + 64 < D_DIM) {
      for (int i = tid; i < 64 * 16; i += 128) {
        int r = i >> 4, c4 = (i & 15) << 2;
        cp16(&ldsAT[nxt][r * 68 + c4], &PHI[(kk + 64 + r) * M_DIM + mblk + c4]);
      }
      for (int i = tid; i < 64 * 4; i += 128) {
        int r = i >> 2, c4 = (i & 3) << 2;
        cp16(&ldsB[nxt][r * 20 + c4], &Res[(kk + 64 + r) * B_DIM + b0 + c4]);
      }
    }
    const float* Ap = ldsAT[cur];
    const float* Bp = ldsB[cur];
#pragma unroll
    for (int k4 = 0; k4 < 64; k4 += 4) {
      int kb = k4 + half * 2;
      v2f a, b;
      a.x = Ap[kb * 68 + wave * 16 + l16];
      a.y = Ap[(kb + 1) * 68 + wave * 16 + l16];
      b.x = Bp[kb * 20 + l16];
      b.y = Bp[(kb + 1) * 20 + l16];
      acc = wmma4(a, b, acc);
    }
    async_fence();
    __syncthreads();
    cur = nxt;
  }

  const float eta = scal[2], etaLam = scal[3], c = coeffs[it];
  const int b = b0 + l16;
#pragma unroll
  for (int i = 0; i < 8; ++i) {
    int m = mblk + wave * 16 + half * 8 + i;
    int idx = m * B_DIM + b;
    float y = fmaf(eta, acc[i], Yv[idx]);       // gradient step
    float an = fmaxf(y - etaLam, 0.f);          // nonneg soft-threshold
    float ao = ahat[idx];
    ahat[idx] = an;
    Yv[idx] = fmaf(an - ao, c, an);             // momentum
  }
}

// ------------- GEMM: new_basis = PHI + (STEP/B) * (Res/freq) * ahat^T / (Hess+eps)
__global__ __launch_bounds__(128) void k_gemm_basis(
    const float* __restrict__ PHI, const float* __restrict__ Res,
    const float* __restrict__ ahat, const float* __restrict__ freq,
    const float* __restrict__ hess, float* __restrict__ nb) {
  __shared__ __align__(16) float ldsA[16 * 132];  // (Res/freq) panel [16 d][128 b]
  const int tid = threadIdx.x;
  const int lane = tid & 31, wave = tid >> 5;
  const int half = lane >> 4, l16 = lane & 15;
  const int dblk = blockIdx.x * 16;
  const int mblk = blockIdx.y * 64;
  for (int i = tid; i < 16 * 32; i += 128) {
    int r = i >> 5, c4 = (i & 31) << 2;
    v4f rv = *(const v4f*)&Res[(dblk + r) * B_DIM + c4];
    v4f fv = *(const v4f*)&freq[c4];
    ldsA[r * 132 + c4 + 0] = rv.x / fv.x; ldsA[r * 132 + c4 + 1] = rv.y / fv.y;
    ldsA[r * 132 + c4 + 2] = rv.z / fv.z; ldsA[r * 132 + c4 + 3] = rv.w / fv.w;
  }
  __syncthreads();
  v8f acc = {};
  const int m = mblk + wave * 16 + l16;
#pragma unroll
  for (int k4 = 0; k4 < B_DIM; k4 += 4) {
    int kb = k4 + half * 2;
    v2f a = *(const v2f*)&ldsA[l16 * 132 + kb];
    v2f b = *(const v2f*)&ahat[m * B_DIM + kb];  // (ahat^T)[b, m] pair
    acc = wmma4(a, b, acc);
  }
  const float coef = (STEP / (float)B_DIM) / (hess[m] + LOWEST);
#pragma unroll
  for (int i = 0; i < 8; ++i) {
    int d = dblk + half * 8 + i;
    nb[d * M_DIM + m] = fmaf(coef, acc[i], PHI[d * M_DIM + m]);
  }
}

// ---------------------------------------------------------- reductions / epilogue
__global__ void k_rowstats(const float* __restrict__ ahat,
                           const float* __restrict__ actin,
                           const float* __restrict__ hessin,
                           float* __restrict__ hessout, float* __restrict__ scal) {
  int m = blockIdx.x * blockDim.x + threadIdx.x;
  if (m >= M_DIM) return;
  float s1 = 0.f, s2 = 0.f;
  for (int b = 0; b < B_DIM; ++b) {
    float a = ahat[m * B_DIM + b];
    s1 += fabsf(a);
    s2 = fmaf(a, a, s2);
  }
  const float decay = (HIST - 1.f) / HIST;
  float act = fmaf(actin[m], decay, s1 / ((float)B_DIM * HIST));
  float h = fmaf(hessin[m], decay, s2 / ((float)B_DIM * HIST));
  hessout[m] = h;
  atomicMax((unsigned*)scal + 6, __float_as_uint(act));
  atomicMin((unsigned*)scal + 7, __float_as_uint(act));
}

__global__ void k_energy(const float* __restrict__ batch,
                         const float* __restrict__ Res, float* __restrict__ scal) {
  __shared__ float red[256];
  int idx = blockIdx.x * 256 + threadIdx.x;  // grid covers D*B exactly
  float bi = batch[idx];
  float ri = Res[idx];
  red[threadIdx.x] = bi * bi;
  __syncthreads();
  for (int s = 128; s > 0; s >>= 1) {
    if (threadIdx.x < s) red[threadIdx.x] += red[threadIdx.x + s];
    __syncthreads();
  }
  if (threadIdx.x == 0) atomicAdd(&scal[4], red[0]);
  __syncthreads();
  red[threadIdx.x] = ri * ri;
  __syncthreads();
  for (int s = 128; s > 0; s >>= 1) {
    if (threadIdx.x < s) red[threadIdx.x] += red[threadIdx.x + s];
    __syncthreads();
  }
  if (threadIdx.x == 0) atomicAdd(&scal[5], red[0]);
}

__global__ void k_colnorm(const float* __restrict__ nb, float* __restrict__ norms) {
  int m = blockIdx.x * blockDim.x + threadIdx.x;
  if (m >= M_DIM) return;
  float s = 0.f;
  for (int d = 0; d < D_DIM; ++d) {
    float v = nb[d * M_DIM + m];
    s = fmaf(v, v, s);
  }
  norms[m] = rsqrtf(s);
}

__global__ void k_colscale(float* __restrict__ nb, const float* __restrict__ norms) {
  int m = blockIdx.x * 256 + threadIdx.x;
  int d = blockIdx.y;
  nb[d * M_DIM + m] *= norms[m];
}

__global__ void k_pack(const float* __restrict__ scal, float* __restrict__ snr,
                       float* __restrict__ mx, float* __restrict__ mn) {
  if (threadIdx.x == 0 && blockIdx.x == 0) {
    *snr = scal[4] / scal[5];
    *mx = __uint_as_float(((const unsigned*)scal)[6]);
    *mn = __uint_as_float(((const unsigned*)scal)[7]);
  }
}

// --------------------------------------------------------------------- launcher
extern "C" void kernel_launch(void* const* d_in, const int* in_sizes, int n_in,
                              void* d_out, int out_size, void* d_ws, size_t ws_size,
                              hipStream_t stream) {
  (void)in_sizes; (void)n_in; (void)out_size; (void)ws_size;
  const float* batch  = (const float*)d_in[0];  // [B, D]
  const float* freq   = (const float*)d_in[1];  // [B]
  const float* PHI    = (const float*)d_in[2];  // [D, M]
  const float* HessIn = (const float*)d_in[3];  // [M]
  const float* ActIn  = (const float*)d_in[4];  // [M]

  float* out = (float*)d_out;
  float* out_recons = out;                                   // [D, B]
  float* out_ahat   = out + D_DIM * B_DIM;                   // [M, B]
  float* out_snr    = out + D_DIM * B_DIM + M_DIM * B_DIM;   // scalar
  float* out_max    = out_snr + 1;
  float* out_min    = out_snr + 2;
  float* out_basis  = out_snr + 3;                           // [D, M]
  float* out_hess   = out_basis + D_DIM * M_DIM;             // [M]

  float* ws     = (float*)d_ws;
  float* Yv     = ws;                          // ahat_y [M, B]
  float* Res    = Yv + M_DIM * B_DIM;          // [D, B]
  float* coeffs = Res + D_DIM * B_DIM;         // [512]
  float* vvec   = coeffs + 512;                // [D]
  float* tvec   = vvec + D_DIM;                // [M]
  float* norms  = tvec + M_DIM;                // [M]
  float* scal   = norms + M_DIM;               // [16]

  k_init<<<1536, 256, 0, stream>>>(Yv, out_ahat, vvec, scal);
  k_coeffs<<<1, 1, 0, stream>>>(coeffs);

  for (int p = 0; p < POWER_ITERS; ++p) {
    k_pk_a<<<12, 256, 0, stream>>>(PHI, vvec, tvec, scal);
    k_pk_b<<<3, 256, 0, stream>>>(PHI, vvec, tvec, scal);
    k_pk_fin<<<1, 768, 0, stream>>>(vvec, scal);
  }

  dim3 gres(D_DIM / 64, B_DIM / 16);    // 12 x 8
  dim3 ggrad(M_DIM / 64, B_DIM / 16);   // 48 x 8
  for (int it = 0; it < NUM_ITER; ++it) {
    k_gemm_res<<<gres, 128, 0, stream>>>(PHI, Yv, batch, Res, nullptr);
    k_gemm_grad<<<ggrad, 128, 0, stream>>>(PHI, Res, out_ahat, Yv, coeffs, scal, it);
  }

  // final residual + reconstruction with ahat
  k_gemm_res<<<gres, 128, 0, stream>>>(PHI, out_ahat, batch, Res, out_recons);
  k_rowstats<<<12, 256, 0, stream>>>(out_ahat, ActIn, HessIn, out_hess, scal);
  k_energy<<<(D_DIM * B_DIM) / 256, 256, 0, stream>>>(batch, Res, scal);

  dim3 gbas(D_DIM / 16, M_DIM / 64);    // 48 x 48
  k_gemm_basis<<<gbas, 128, 0, stream>>>(PHI, Res, out_ahat, freq, out_hess, out_basis);
  k_colnorm<<<12, 256, 0, stream>>>(out_basis, norms);
  dim3 gscale(M_DIM / 256, D_DIM);      // 12 x 768
  k_colscale<<<gscale, 256, 0, stream>>>(out_basis, norms);
  k_pack<<<1, 1, 0, stream>>>(scal, out_snr, out_max, out_min);
}